// WaveletTokenizer_14740327760386
// MI455X (gfx1250) — compile-verified
//
#include <hip/hip_runtime.h>
#include <math.h>

typedef __attribute__((ext_vector_type(2))) float v2f;
typedef __attribute__((ext_vector_type(8))) float v8f;

#define EPSF 1e-8f

// Problem sizes (fixed by the reference setup_inputs)
#define BATCH 2
#define L0 73728   // 64*64*6*3
#define L1 18432   // 32*32*6*3
#define L2 4608    // 16*16*6*3
#define L3 768     // 16*16*1*3
#define OFF1 73728
#define OFF2 92160
#define OFF3 96768
#define LTOT 97536
#define VCB 4096
#define TILES_PER_BATCH (LTOT / 16)       // 6096
#define NTILES (BATCH * TILES_PER_BATCH)  // 12192

// ---------------- zero accumulators ----------------
__global__ void zero_kernel(float* __restrict__ lvl_sums) {
    if (threadIdx.x < 4) lvl_sums[threadIdx.x] = 0.f;
}

// ---------------- per-(level,batch) amp statistics ----------------
__global__ __launch_bounds__(256) void stats_kernel(
    const float* __restrict__ yh0, const float* __restrict__ yh1,
    const float* __restrict__ yh2, const float* __restrict__ yl,
    float* __restrict__ stats)
{
    __shared__ float ssum[256];
    __shared__ float ssq[256];
    int pair = blockIdx.x;       // pair = lvl*2 + b, 8 pairs
    int lvl = pair >> 1;
    int b   = pair & 1;
    const float* src; int n; int cplx;
    if      (lvl == 0) { src = yh0; n = L0; cplx = 1; }
    else if (lvl == 1) { src = yh1; n = L1; cplx = 1; }
    else if (lvl == 2) { src = yh2; n = L2; cplx = 1; }
    else               { src = yl;  n = L3; cplx = 0; }

    float s = 0.f, sq = 0.f;
    for (int i = threadIdx.x; i < n; i += 256) {
        float a;
        if (cplx) {
            size_t ix = (size_t)(b * n + i) * 2;
            float re = src[ix], im = src[ix + 1];
            a = sqrtf(re * re + im * im);
        } else {
            a = fabsf(src[b * n + i]);
        }
        s += a; sq += a * a;
    }
    ssum[threadIdx.x] = s; ssq[threadIdx.x] = sq;
    __syncthreads();
    for (int off = 128; off > 0; off >>= 1) {
        if ((int)threadIdx.x < off) {
            ssum[threadIdx.x] += ssum[threadIdx.x + off];
            ssq[threadIdx.x]  += ssq[threadIdx.x + off];
        }
        __syncthreads();
    }
    if (threadIdx.x == 0) {
        float mean = ssum[0] / (float)n;
        float var  = (ssq[0] - (float)n * mean * mean) / (float)(n - 1);
        float sd   = sqrtf(fmaxf(var, 0.f)) + EPSF;   // torch-style ddof=1 + EPS
        stats[pair * 2 + 0] = mean;
        stats[pair * 2 + 1] = sd;
    }
}

// ---------------- tokenization: write (B, LTOT, 3) tokens ----------------
__global__ __launch_bounds__(256) void tokenize_kernel(
    const float* __restrict__ yh0, const float* __restrict__ yh1,
    const float* __restrict__ yh2, const float* __restrict__ yl,
    const float* __restrict__ level_scale, const float* __restrict__ stats,
    float* __restrict__ tokens)
{
    int g = blockIdx.x * 256 + threadIdx.x;
    if (g >= BATCH * LTOT) return;
    int b = g / LTOT;
    int t = g % LTOT;
    int lvl, local, HW;
    const float* src;
    if      (t < OFF1) { lvl = 0; local = t;        src = yh0; HW = 64; }
    else if (t < OFF2) { lvl = 1; local = t - OFF1; src = yh1; HW = 32; }
    else if (t < OFF3) { lvl = 2; local = t - OFF2; src = yh2; HW = 16; }
    else               { lvl = 3; local = t - OFF3; src = yl;  HW = 16; }
    float mean = stats[(lvl * 2 + b) * 2 + 0];
    float sd   = stats[(lvl * 2 + b) * 2 + 1];
    float sc   = level_scale[lvl];
    float ampn, cs, sn;
    if (lvl < 3) {
        // token order within batch: ((h*W + w)*O + o)*C + c ; C=3, O=6
        int c = local % 3;
        int o = (local / 3) % 6;
        int w = (local / 18) % HW;
        int h = local / (18 * HW);
        size_t si = ((((size_t)(b * 3 + c) * 6 + o) * HW + h) * HW + w) * 2;
        float re = src[si], im = src[si + 1];
        float amp = sqrtf(re * re + im * im);
        ampn = (amp - mean) / sd * sc;
        cs = re / (amp + EPSF);
        sn = im / (amp + EPSF);
    } else {
        // token order: (h*W + w)*C + c ; C=3, H=W=16
        int c = local % 3;
        int w = (local / 3) % 16;
        int h = local / 48;
        size_t si = (((size_t)(b * 3 + c) * 16 + h) * 16 + w);
        float v = src[si];
        float amp = fabsf(v);
        ampn = (amp - mean) / sd * sc;
        cs = v / (amp + EPSF);
        sn = 0.f;
    }
    size_t o3 = (size_t)g * 3;
    tokens[o3 + 0] = ampn;
    tokens[o3 + 1] = cs;
    tokens[o3 + 2] = sn;
}

// ---------------- WMMA VQ argmin ----------------
// Score S[m][n] = -2*f_m . e_n + |e_n|^2  (same argmin as full L2 distance).
// A (16x4): row m = (-2fx, -2fy, -2fz, 1)    B (4x16): col n = (ex, ey, ez, |e|^2)
__global__ __launch_bounds__(256) void vq_kernel(
    const float* __restrict__ tokens, const float* __restrict__ E,
    float* __restrict__ quant_out, float* __restrict__ idx_out,
    float* __restrict__ lvl_sums)
{
    __shared__ float4 sE[VCB];   // 64 KB: (ex, ey, ez, |e|^2)
    int tid = threadIdx.x;
    for (int j = tid; j < VCB; j += 256) {
        float ex = E[j * 3 + 0], ey = E[j * 3 + 1], ez = E[j * 3 + 2];
        sE[j] = make_float4(ex, ey, ez, ex * ex + ey * ey + ez * ez);
    }
    __syncthreads();

    int wave = tid >> 5;         // 8 waves per block, one 16-token tile each
    int lane = tid & 31;
    int p  = lane & 15;
    int hi = lane >> 4;

    int tile = blockIdx.x * 8 + wave;
    if (tile >= NTILES) return;                 // wave-uniform
    int b    = tile / TILES_PER_BATCH;
    int tpos = (tile % TILES_PER_BATCH) * 16;   // token offset within batch
    size_t base = ((size_t)b * LTOT + tpos) * 3;

    // each lane loads token m = p (both half-groups load all 16 tokens)
    float fx = tokens[base + p * 3 + 0];
    float fy = tokens[base + p * 3 + 1];
    float fz = tokens[base + p * 3 + 2];
    float fsq = fx * fx + fy * fy + fz * fz;

    // A fragment (16x4 f32): VGPR0 = K0(lanes0-15)/K2(lanes16-31), VGPR1 = K1/K3
    v2f a;
    if (hi == 0) { a.x = -2.f * fx; a.y = -2.f * fy; }
    else         { a.x = -2.f * fz; a.y = 1.0f;      }

    float minv[8];
    int   mini[8];
#pragma unroll
    for (int r = 0; r < 8; ++r) { minv[r] = 3.4e38f; mini[r] = 0; }

    // software-pipelined, 2x-unrolled: prefetch next pair of B fragments
    // (wrapped index keeps the prefetch branch-free; the redundant final
    // loads are harmless LDS reads)
    float4 e0 = sE[p];
    float4 e1 = sE[16 + p];
    for (int c0 = 0; c0 < VCB; c0 += 32) {
        v2f bf0, bf1;
        bf0.x = hi ? e0.z : e0.x;  bf0.y = hi ? e0.w : e0.y;
        bf1.x = hi ? e1.z : e1.x;  bf1.y = hi ? e1.w : e1.y;
        int nc = (c0 + 32) & (VCB - 1);
        e0 = sE[nc + p];
        e1 = sE[nc + 16 + p];

        v8f acc0 = {}, acc1 = {};
        acc0 = __builtin_amdgcn_wmma_f32_16x16x4_f32(
            false, a, false, bf0, (short)0, acc0, false, false);
        acc1 = __builtin_amdgcn_wmma_f32_16x16x4_f32(
            false, a, false, bf1, (short)0, acc1, false, false);

        int n0 = c0 + p;
        int n1 = c0 + 16 + p;
#pragma unroll
        for (int r = 0; r < 8; ++r) {
            float s0 = acc0[r];
            float s1 = acc1[r];
            // strict < keeps the first (lowest-index) minimum per lane
            bool t0 = s0 < minv[r];
            minv[r] = t0 ? s0 : minv[r];
            mini[r] = t0 ? n0 : mini[r];
            bool t1 = s1 < minv[r];
            minv[r] = t1 ? s1 : minv[r];
            mini[r] = t1 ? n1 : mini[r];
        }
    }

    // branchless argmin reduction across the 16 lanes of each half-group:
    // phase 1: fmin-reduce the value; phase 2: min-reduce the index among
    // lanes holding the group minimum (matches argmin-first tie-break).
#pragma unroll
    for (int r = 0; r < 8; ++r) {
        float gmin = minv[r];
        gmin = fminf(gmin, __shfl_xor(gmin, 8, 16));
        gmin = fminf(gmin, __shfl_xor(gmin, 4, 16));
        gmin = fminf(gmin, __shfl_xor(gmin, 2, 16));
        gmin = fminf(gmin, __shfl_xor(gmin, 1, 16));
        int cand = (minv[r] == gmin) ? mini[r] : 0x7FFFFFFF;
        cand = min(cand, __shfl_xor(cand, 8, 16));
        cand = min(cand, __shfl_xor(cand, 4, 16));
        cand = min(cand, __shfl_xor(cand, 2, 16));
        cand = min(cand, __shfl_xor(cand, 1, 16));
        minv[r] = gmin;
        mini[r] = cand;
    }

    // hi==0 lanes 0..7 write rows m=0..7 ; hi==1 lanes p=8..15 write rows m=8..15
    bool writer = (hi == 0) ? (p < 8) : (p >= 8);
    if (writer) {
        int m = p;
        int r = hi ? (p - 8) : p;
        int idx = mini[r];
        float dmin = minv[r] + fsq;     // |f|^2 - 2 f.e + |e|^2
        size_t tok = (size_t)b * LTOT + tpos + m;
        idx_out[tok] = (float)idx;
        float4 e = sE[idx];
        quant_out[tok * 3 + 0] = e.x;
        quant_out[tok * 3 + 1] = e.y;
        quant_out[tok * 3 + 2] = e.z;
        int lvl = (tpos < OFF1) ? 0 : (tpos < OFF2 ? 1 : (tpos < OFF3 ? 2 : 3));
        atomicAdd(&lvl_sums[lvl], dmin);
    }
}

// ---------------- final loss ----------------
__global__ void loss_kernel(const float* __restrict__ lvl_sums, float* __restrict__ out_loss) {
    float l = 0.f;
    l += 1.25f * lvl_sums[0] / (float)(BATCH * L0 * 3);
    l += 1.25f * lvl_sums[1] / (float)(BATCH * L1 * 3);
    l += 1.25f * lvl_sums[2] / (float)(BATCH * L2 * 3);
    l += 1.25f * lvl_sums[3] / (float)(BATCH * L3 * 3);
    *out_loss = l;
}

extern "C" void kernel_launch(void* const* d_in, const int* in_sizes, int n_in,
                              void* d_out, int out_size, void* d_ws, size_t ws_size,
                              hipStream_t stream) {
    const float* yh0 = (const float*)d_in[0];
    const float* yh1 = (const float*)d_in[1];
    const float* yh2 = (const float*)d_in[2];
    const float* yl  = (const float*)d_in[3];
    const float* emb = (const float*)d_in[4];
    const float* lsc = (const float*)d_in[5];

    float* out   = (float*)d_out;
    float* quant = out;                                  // BATCH*LTOT*3
    float* idxf  = out + (size_t)BATCH * LTOT * 3;       // BATCH*LTOT
    float* lossp = idxf + (size_t)BATCH * LTOT;          // 1

    float* wsf      = (float*)d_ws;
    float* tokens   = wsf;                               // BATCH*LTOT*3 = 585216
    float* stats    = wsf + (size_t)BATCH * LTOT * 3;    // 16 floats
    float* lvl_sums = stats + 16;                        // 4 floats

    zero_kernel<<<1, 32, 0, stream>>>(lvl_sums);
    stats_kernel<<<8, 256, 0, stream>>>(yh0, yh1, yh2, yl, stats);
    int ntok = BATCH * LTOT;
    tokenize_kernel<<<(ntok + 255) / 256, 256, 0, stream>>>(
        yh0, yh1, yh2, yl, lsc, stats, tokens);
    vq_kernel<<<NTILES / 8, 256, 0, stream>>>(tokens, emb, quant, idxf, lvl_sums);
    loss_kernel<<<1, 1, 0, stream>>>(lvl_sums, lossp);
}